// LocalCrossModalAttention_28398323761482
// MI455X (gfx1250) — compile-verified
//
#include <hip/hip_runtime.h>
#include <hip/hip_bf16.h>
#include <math.h>

// ---------------------------------------------------------------------------
// Sizes (compile-time, match reference)
// ---------------------------------------------------------------------------
#define L_TOK 3072
#define GEO_D 1536
#define SEM_D 512
#define RSA_D 64
#define D_MOD 256
#define N_HEAD 8
#define HD_DIM 32
#define KNN_K 10
#define LN_EPS 1e-5f

typedef __attribute__((ext_vector_type(2))) float v2f;
typedef __attribute__((ext_vector_type(8))) float v8f;

#define TILE_M 16
#define BK 32            // K panel depth staged in LDS
#define BKP (BK + 2)     // row pad: 8B alignment for ds_load_b64 + bank spread

// ---------------------------------------------------------------------------
// fp32 WMMA GEMM with fused epilogue, templated on tiles-per-wave (N=MYT*128)
// so every loop is compile-time and the WMMA chain is branch-free.
//   pre  = X @ W + bias
//   if res:   pre = alpha*res + beta*pre
//   if do_ln: pre = LN(pre)*gamma + lnb
//   if leaky: pre = leakyrelu(pre, 0.01)
// Block = 16 rows x N cols, 8 waves; A/B fragments via ds_load_b64; B panel
// staged transposed [n][k] with float4 global loads; LN buffer overlays sB.
// ---------------------------------------------------------------------------
template <int MYT>
__global__ __launch_bounds__(256)
void gemm_epilogue_kernel(const float* __restrict__ X, int ldx,
                          const float* __restrict__ W,
                          const float* __restrict__ bias,
                          int Kd,
                          const float* __restrict__ res,
                          const float* __restrict__ alpha_p,
                          const float* __restrict__ beta_p,
                          const float* __restrict__ gamma,
                          const float* __restrict__ lnb,
                          int do_ln, int do_leaky,
                          float* __restrict__ out, int ldo, int col_off)
{
    constexpr int N = MYT * 128;

    __shared__ float smem[N * BKP];          // sB (transposed W panel) / sOut overlay
    __shared__ float sA[TILE_M * BKP];       // A panel
    __shared__ float sSum[TILE_M][16], sSum2[TILE_M][16];
    __shared__ float sMean[TILE_M], sRstd[TILE_M];

    float* sB   = smem;                      // [N][BKP] : sB[n*BKP + k] = W[kb+k][n]
    float* sOut = smem;                      // [TILE_M][N] (after final barrier)

    const int tid  = threadIdx.x;
    const int wave = tid >> 5;
    const int lane = tid & 31;
    const int row0 = blockIdx.x * TILE_M;

    const int lm    = lane & 15;             // column within 16-wide tile / A row
    const int khalf = (lane >> 4) << 1;      // 0 (lanes 0-15) or 2 (lanes 16-31)

    // staging coordinates (compile-time trip counts, no guards)
    const int ar = tid >> 4;                 // A row 0..15
    const int ac = (tid & 15) << 1;          // A col pair 0,2,..,30

    int n0[MYT];
    #pragma unroll
    for (int t = 0; t < MYT; ++t) n0[t] = (wave + (t << 3)) << 4;

    v8f acc[MYT];
    #pragma unroll
    for (int t = 0; t < MYT; ++t) acc[t] = v8f{};

    for (int kb = 0; kb < Kd; kb += BK) {
        // ---- A panel: one float2 load + one b64 LDS store per thread ----
        {
            const float2 x2 = *reinterpret_cast<const float2*>(
                &X[(size_t)(row0 + ar) * ldx + kb + ac]);
            v2f a2; a2.x = x2.x; a2.y = x2.y;
            *reinterpret_cast<v2f*>(&sA[ar * BKP + ac]) = a2;
        }
        // ---- B panel: float4 coalesced loads, transposed LDS stores ----
        #pragma unroll
        for (int j = 0; j < (BK * N) / 1024; ++j) {
            const int i = (tid + j * 256) << 2;       // 4 consecutive elements
            const int k = i / N, n = i - k * N;       // N pow2 -> shifts
            const float4 w4 = *reinterpret_cast<const float4*>(
                &W[(size_t)(kb + k) * N + n]);
            sB[(n + 0) * BKP + k] = w4.x;
            sB[(n + 1) * BKP + k] = w4.y;
            sB[(n + 2) * BKP + k] = w4.z;
            sB[(n + 3) * BKP + k] = w4.w;
        }
        __syncthreads();

        // ---- branch-free WMMA chain: 8 k-steps x MYT tiles ----
        #pragma unroll
        for (int kk = 0; kk < BK; kk += 4) {
            const v2f a = *reinterpret_cast<const v2f*>(&sA[lm * BKP + kk + khalf]);
            #pragma unroll
            for (int t = 0; t < MYT; ++t) {
                const v2f b = *reinterpret_cast<const v2f*>(
                    &sB[(n0[t] + lm) * BKP + kk + khalf]);
                acc[t] = __builtin_amdgcn_wmma_f32_16x16x4_f32(
                    false, a, false, b, (short)0, acc[t], false, false);
            }
        }
        __syncthreads();   // also guards the sOut overlay after the last block
    }

    const float alpha = alpha_p ? *alpha_p : 1.0f;
    const float beta  = beta_p  ? *beta_p  : 1.0f;

    if (!do_ln) {
        // Fast path (Q/K/V projections): accumulators straight to global.
        #pragma unroll
        for (int t = 0; t < MYT; ++t) {
            const int col = n0[t] + lm;
            const float bv = bias ? bias[col] : 0.0f;
            #pragma unroll
            for (int r = 0; r < 8; ++r) {
                const int mrow = (lane < 16) ? r : (r + 8);
                float v = acc[t][r] + bv;
                if (res) v = alpha * res[(size_t)(row0 + mrow) * N + col] + beta * v;
                if (do_leaky) v = (v >= 0.f) ? v : 0.01f * v;
                out[(size_t)(row0 + mrow) * ldo + col_off + col] = v;
            }
        }
        return;
    }

    // ---- LN path: stage tile (bias + scaled residual), row stats, write ----
    #pragma unroll
    for (int t = 0; t < MYT; ++t) {
        const int col = n0[t] + lm;
        const float bv = bias ? bias[col] : 0.0f;
        #pragma unroll
        for (int r = 0; r < 8; ++r) {
            const int mrow = (lane < 16) ? r : (r + 8);
            float v = acc[t][r] + bv;
            if (res) v = alpha * res[(size_t)(row0 + mrow) * N + col] + beta * v;
            sOut[mrow * N + col] = v;
        }
    }
    __syncthreads();

    // parallel row statistics: 16 threads per row
    {
        const int r = tid >> 4, j = tid & 15;
        float s = 0.f, s2 = 0.f;
        #pragma unroll
        for (int c = j; c < N; c += 16) {
            const float x = sOut[r * N + c]; s += x; s2 += x * x;
        }
        sSum[r][j] = s; sSum2[r][j] = s2;
    }
    __syncthreads();
    if (tid < TILE_M) {
        float s = 0.f, s2 = 0.f;
        #pragma unroll
        for (int j = 0; j < 16; ++j) { s += sSum[tid][j]; s2 += sSum2[tid][j]; }
        const float m   = s / (float)N;
        const float var = s2 / (float)N - m * m;
        sMean[tid] = m;
        sRstd[tid] = rsqrtf(var + LN_EPS);
    }
    __syncthreads();

    #pragma unroll
    for (int j = 0; j < (TILE_M * N) / 256; ++j) {
        const int i = tid + j * 256;
        const int r = i >> (MYT == 2 ? 8 : 9);     // i / N
        const int c = i & (N - 1);
        float v = (sOut[r * N + c] - sMean[r]) * sRstd[r] * gamma[c] + lnb[c];
        if (do_leaky) v = (v >= 0.f) ? v : 0.01f * v;
        out[(size_t)(row0 + r) * ldo + col_off + c] = v;
    }
}

// ---------------------------------------------------------------------------
// 10-NN by squared euclidean distance. One wave32 per query row; each lane
// keeps a sorted local top-10 over a strided slice, lane0 merges the 32
// sorted lists. Tie-break: smaller index (matches jax.lax.top_k).
// ---------------------------------------------------------------------------
__global__ __launch_bounds__(32)
void knn_kernel(const float* __restrict__ pos, int* __restrict__ knn)
{
    __shared__ float sd[32 * KNN_K];
    __shared__ int   si[32 * KNN_K];
    const int i    = blockIdx.x;
    const int lane = threadIdx.x;

    const float px = pos[i * 3 + 0], py = pos[i * 3 + 1], pz = pos[i * 3 + 2];

    float bd[KNN_K]; int bi[KNN_K];
    #pragma unroll
    for (int t = 0; t < KNN_K; ++t) { bd[t] = INFINITY; bi[t] = 0x7fffffff; }

    for (int j = lane; j < L_TOK; j += 32) {
        const float dx = pos[j * 3 + 0] - px;
        const float dy = pos[j * 3 + 1] - py;
        const float dz = pos[j * 3 + 2] - pz;
        const float d2 = dx * dx + dy * dy + dz * dz;
        if (d2 < bd[KNN_K - 1] || (d2 == bd[KNN_K - 1] && j < bi[KNN_K - 1])) {
            int t = KNN_K - 1;
            while (t > 0 && (d2 < bd[t - 1] || (d2 == bd[t - 1] && j < bi[t - 1]))) {
                bd[t] = bd[t - 1]; bi[t] = bi[t - 1]; --t;
            }
            bd[t] = d2; bi[t] = j;
        }
    }
    for (int t = 0; t < KNN_K; ++t) { sd[lane * KNN_K + t] = bd[t]; si[lane * KNN_K + t] = bi[t]; }
    __syncthreads();

    if (lane == 0) {
        int ptr[32];
        for (int w = 0; w < 32; ++w) ptr[w] = 0;
        for (int t = 0; t < KNN_K; ++t) {
            float best = INFINITY; int bw = 0, bidx = 0x7fffffff;
            for (int w = 0; w < 32; ++w) {
                if (ptr[w] < KNN_K) {
                    const float d  = sd[w * KNN_K + ptr[w]];
                    const int   id = si[w * KNN_K + ptr[w]];
                    if (d < best || (d == best && id < bidx)) { best = d; bw = w; bidx = id; }
                }
            }
            knn[i * KNN_K + t] = bidx;
            ++ptr[bw];
        }
    }
}

// ---------------------------------------------------------------------------
// Sparse k-NN attention: softmax over exactly the 10 neighbor scores.
// One block per query row; wave w == head w (wave32 == 32 head dims).
// ---------------------------------------------------------------------------
__global__ __launch_bounds__(256)
void sparse_attn_kernel(const float* __restrict__ Q, const float* __restrict__ Kb,
                        const float* __restrict__ V, const int* __restrict__ knn,
                        float* __restrict__ Outp)
{
    const int i    = blockIdx.x;
    const int head = threadIdx.x >> 5;
    const int lane = threadIdx.x & 31;
    const int dcol = head * HD_DIM + lane;

    const float q = Q[(size_t)i * D_MOD + dcol];

    int idxs[KNN_K];
    #pragma unroll
    for (int t = 0; t < KNN_K; ++t) idxs[t] = knn[i * KNN_K + t];

    float s[KNN_K];
    float mx = -INFINITY;
    #pragma unroll
    for (int t = 0; t < KNN_K; ++t) {
        float p = q * Kb[(size_t)idxs[t] * D_MOD + dcol];
        #pragma unroll
        for (int off = 16; off > 0; off >>= 1) p += __shfl_xor(p, off, 32);
        p *= 0.17677669529663687f;   // 1/sqrt(HD)
        s[t] = p;
        mx = fmaxf(mx, p);
    }
    float sum = 0.f;
    #pragma unroll
    for (int t = 0; t < KNN_K; ++t) { s[t] = expf(s[t] - mx); sum += s[t]; }
    const float inv = 1.0f / sum;

    float o = 0.f;
    #pragma unroll
    for (int t = 0; t < KNN_K; ++t) o += s[t] * V[(size_t)idxs[t] * D_MOD + dcol];
    Outp[(size_t)i * D_MOD + dcol] = o * inv;
}

// ---------------------------------------------------------------------------
// Host orchestration
// ---------------------------------------------------------------------------
extern "C" void kernel_launch(void* const* d_in, const int* in_sizes, int n_in,
                              void* d_out, int out_size, void* d_ws, size_t ws_size,
                              hipStream_t stream)
{
    (void)in_sizes; (void)n_in; (void)out_size; (void)ws_size;

    const float* geo = (const float*)d_in[0];
    const float* sem = (const float*)d_in[1];
    const float* rsa = (const float*)d_in[2];
    const float* pos = (const float*)d_in[3];
    const float* Wg  = (const float*)d_in[4];  const float* bg  = (const float*)d_in[5];
    const float* gg  = (const float*)d_in[6];  const float* bgn = (const float*)d_in[7];
    const float* Ws  = (const float*)d_in[8];  const float* bs  = (const float*)d_in[9];
    const float* gs  = (const float*)d_in[10]; const float* bsn = (const float*)d_in[11];
    const float* Wr  = (const float*)d_in[12]; const float* br  = (const float*)d_in[13];
    const float* gr  = (const float*)d_in[14]; const float* brn = (const float*)d_in[15];
    const float* Wq  = (const float*)d_in[16]; const float* bq  = (const float*)d_in[17];
    const float* Wk  = (const float*)d_in[18]; const float* bk  = (const float*)d_in[19];
    const float* Wv  = (const float*)d_in[20]; const float* bv  = (const float*)d_in[21];
    const float* Wo  = (const float*)d_in[22]; const float* bo  = (const float*)d_in[23];
    const float* a_g = (const float*)d_in[24]; const float* be_g= (const float*)d_in[25];
    const float* a_s = (const float*)d_in[26]; const float* be_s= (const float*)d_in[27];
    const float* Wt  = (const float*)d_in[28]; const float* bt  = (const float*)d_in[29];
    const float* gt  = (const float*)d_in[30]; const float* btn = (const float*)d_in[31];
    const float* Wf1 = (const float*)d_in[32]; const float* bf1 = (const float*)d_in[33];
    const float* gf1 = (const float*)d_in[34]; const float* bf1n= (const float*)d_in[35];
    const float* Wf2 = (const float*)d_in[36]; const float* bf2 = (const float*)d_in[37];
    const float* gf2 = (const float*)d_in[38]; const float* bf2n= (const float*)d_in[39];
    const float* g1  = (const float*)d_in[40]; const float* b1  = (const float*)d_in[41];
    const float* g2  = (const float*)d_in[42]; const float* b2  = (const float*)d_in[43];

    float* ws = (float*)d_ws;
    const size_t LD = (size_t)L_TOK * D_MOD;
    float* geo_p    = ws;
    float* sem_p    = ws + 1 * LD;
    float* rsa_p    = ws + 2 * LD;
    float* bufQ     = ws + 3 * LD;
    float* bufK     = ws + 4 * LD;
    float* bufV     = ws + 5 * LD;
    float* bufA     = ws + 6 * LD;
    float* combined = ws + 7 * LD;                    // [L, 768]
    float* bufH     = combined + (size_t)L_TOK * 768; // [L, 512]
    int*   knn      = (int*)(bufH + (size_t)L_TOK * 512);

    const dim3 gGemm(L_TOK / TILE_M), bGemm(256);
    const dim3 gRow(L_TOK);

    // 1) k-NN index
    knn_kernel<<<gRow, dim3(32), 0, stream>>>(pos, knn);

    // 2) modality projections + LN
    gemm_epilogue_kernel<2><<<gGemm, bGemm, 0, stream>>>(geo, GEO_D, Wg, bg, GEO_D,
        nullptr, nullptr, nullptr, gg, bgn, 1, 0, geo_p, D_MOD, 0);
    gemm_epilogue_kernel<2><<<gGemm, bGemm, 0, stream>>>(sem, SEM_D, Ws, bs, SEM_D,
        nullptr, nullptr, nullptr, gs, bsn, 1, 0, sem_p, D_MOD, 0);
    gemm_epilogue_kernel<2><<<gGemm, bGemm, 0, stream>>>(rsa, RSA_D, Wr, br, RSA_D,
        nullptr, nullptr, nullptr, gr, brn, 1, 0, rsa_p, D_MOD, 0);

    // 3) attention 1: q=geo, k/v=sem
    gemm_epilogue_kernel<2><<<gGemm, bGemm, 0, stream>>>(geo_p, D_MOD, Wq, bq, D_MOD,
        nullptr, nullptr, nullptr, nullptr, nullptr, 0, 0, bufQ, D_MOD, 0);
    gemm_epilogue_kernel<2><<<gGemm, bGemm, 0, stream>>>(sem_p, D_MOD, Wk, bk, D_MOD,
        nullptr, nullptr, nullptr, nullptr, nullptr, 0, 0, bufK, D_MOD, 0);
    gemm_epilogue_kernel<2><<<gGemm, bGemm, 0, stream>>>(sem_p, D_MOD, Wv, bv, D_MOD,
        nullptr, nullptr, nullptr, nullptr, nullptr, 0, 0, bufV, D_MOD, 0);
    sparse_attn_kernel<<<gRow, dim3(256), 0, stream>>>(bufQ, bufK, bufV, knn, bufA);
    gemm_epilogue_kernel<2><<<gGemm, bGemm, 0, stream>>>(bufA, D_MOD, Wo, bo, D_MOD,
        geo_p, a_g, be_g, g1, b1, 1, 0, combined, 768, 0);

    // 4) attention 2: q=sem, k/v=geo
    gemm_epilogue_kernel<2><<<gGemm, bGemm, 0, stream>>>(sem_p, D_MOD, Wq, bq, D_MOD,
        nullptr, nullptr, nullptr, nullptr, nullptr, 0, 0, bufQ, D_MOD, 0);
    gemm_epilogue_kernel<2><<<gGemm, bGemm, 0, stream>>>(geo_p, D_MOD, Wk, bk, D_MOD,
        nullptr, nullptr, nullptr, nullptr, nullptr, 0, 0, bufK, D_MOD, 0);
    gemm_epilogue_kernel<2><<<gGemm, bGemm, 0, stream>>>(geo_p, D_MOD, Wv, bv, D_MOD,
        nullptr, nullptr, nullptr, nullptr, nullptr, 0, 0, bufV, D_MOD, 0);
    sparse_attn_kernel<<<gRow, dim3(256), 0, stream>>>(bufQ, bufK, bufV, knn, bufA);
    gemm_epilogue_kernel<2><<<gGemm, bGemm, 0, stream>>>(bufA, D_MOD, Wo, bo, D_MOD,
        sem_p, a_s, be_s, g2, b2, 1, 0, combined, 768, D_MOD);

    // 5) rsa branch -> combined[:, 512:768]
    gemm_epilogue_kernel<2><<<gGemm, bGemm, 0, stream>>>(rsa_p, D_MOD, Wt, bt, D_MOD,
        nullptr, nullptr, nullptr, gt, btn, 1, 1, combined, 768, 2 * D_MOD);

    // 6) fusion FFN
    gemm_epilogue_kernel<4><<<gGemm, bGemm, 0, stream>>>(combined, 768, Wf1, bf1, 768,
        nullptr, nullptr, nullptr, gf1, bf1n, 1, 1, bufH, 512, 0);
    gemm_epilogue_kernel<2><<<gGemm, bGemm, 0, stream>>>(bufH, 512, Wf2, bf2, 512,
        nullptr, nullptr, nullptr, gf2, bf2n, 1, 1, (float*)d_out, D_MOD, 0);
}